// MaskingModule_39968965657010
// MI455X (gfx1250) — compile-verified
//
#include <hip/hip_runtime.h>

typedef __attribute__((ext_vector_type(2))) float v2f;
typedef __attribute__((ext_vector_type(8))) float v8f;

#define NBATCH   4
#define LSEQ     512
#define DDIM     1024
#define PPAT     768
#define NBINS    64
#define LEN_KEEP 128
#define EPSV     1e-19f
#define SIG_INV  100.0f            // 1/sigma, sigma = 0.01

#define OUT_XMASK_OFF 0            // 4*128*1024 = 524288 floats
#define OUT_MASK_OFF  524288       // 4*512 floats
#define OUT_IDSR_OFF  526336       // 4*512 floats (ids_restore as float)

// ---------------- pass 1: per-block min/max partials ----------------
__global__ void __launch_bounds__(256) k_minmax_part(const float* __restrict__ img,
                                                     float* __restrict__ wsf) {
    const int total = NBATCH * LSEQ * PPAT;
    int t = threadIdx.x;
    float mn = 3.4e38f, mx = -3.4e38f;
    for (int i = blockIdx.x * 256 + t; i < total; i += gridDim.x * 256) {
        float v = img[i];
        mn = fminf(mn, v);
        mx = fmaxf(mx, v);
    }
    __shared__ float smn[256], smx[256];
    smn[t] = mn; smx[t] = mx;
    __syncthreads();
    for (int s = 128; s > 0; s >>= 1) {
        if (t < s) { smn[t] = fminf(smn[t], smn[t + s]); smx[t] = fmaxf(smx[t], smx[t + s]); }
        __syncthreads();
    }
    if (t == 0) { wsf[blockIdx.x] = smn[0]; wsf[256 + blockIdx.x] = smx[0]; }
}

// ---------------- pass 2: final min/max -> wsf[512], wsf[513] ----------------
__global__ void __launch_bounds__(256) k_minmax_final(float* __restrict__ wsf, int npart) {
    int t = threadIdx.x;
    __shared__ float smn[256], smx[256];
    smn[t] = (t < npart) ? wsf[t]       :  3.4e38f;
    smx[t] = (t < npart) ? wsf[256 + t] : -3.4e38f;
    __syncthreads();
    for (int s = 128; s > 0; s >>= 1) {
        if (t < s) { smn[t] = fminf(smn[t], smn[t + s]); smx[t] = fmaxf(smx[t], smx[t + s]); }
        __syncthreads();
    }
    if (t == 0) { wsf[512] = smn[0]; wsf[513] = smx[0]; }
}

// ---------------- KDE entropy: one block per 16 (n,l) rows ----------------
// Stage A: stage 16x768 normalized values into LDS
// Stage B: 256 threads compute pdf[16][64] (768-long exp sums each, 4 per thread)
// Stage C: wave 0 reduces over the bin axis twice with chained
//          V_WMMA_F32_16X16X4_F32 (A = pdf tile, B = ones) for
//          (1) the pdf normalization denominator, (2) the entropy sum.
__global__ void __launch_bounds__(256) k_entropy(const float* __restrict__ img,
                                                 float* __restrict__ wsf) {
    __shared__ float s_nv[16 * PPAT];     // 48 KB
    __shared__ float s_pdf[16 * NBINS];   //  4 KB
    __shared__ float s_den[16];

    int t = threadIdx.x;
    float vmin  = wsf[512];
    float vmax  = wsf[513];
    float scale = 1.0f / (vmax - vmin);

    size_t base = (size_t)blockIdx.x * (16 * PPAT);
    for (int i = t; i < 16 * PPAT; i += 256)
        s_nv[i] = (img[base + i] - vmin) * scale;
    __syncthreads();

    // pdf[row][bin] = mean_p exp(-0.5*((nv - bin/63)*100)^2)
    for (int k = 0; k < 4; ++k) {
        int pair = t + k * 256;           // 1024 (row,bin) pairs
        int row  = pair >> 6;
        int bin  = pair & 63;
        float bv = (float)bin * (1.0f / 63.0f);
        const float* rp = &s_nv[row * PPAT];
        float acc = 0.0f;
        #pragma unroll 4
        for (int p = 0; p < PPAT; ++p) {
            float r = (rp[p] - bv) * SIG_INV;
            acc += __expf(-0.5f * r * r);
        }
        s_pdf[row * NBINS + bin] = acc * (1.0f / (float)PPAT);
    }
    __syncthreads();

    if (t < 32) {                         // single full wave: EXEC all ones
        int lane = t;
        int m    = lane & 15;             // A-matrix row this lane supplies
        int kh   = (lane >> 4) << 1;      // K base: 0 for lanes 0-15, 2 for 16-31
        v2f ones; ones.x = 1.0f; ones.y = 1.0f;

        // --- denominator: rowsum_b pdf[m][b] via 16 accumulating WMMAs ---
        v8f c = {};
        for (int ch = 0; ch < 16; ++ch) {
            v2f a;
            a.x = s_pdf[m * NBINS + ch * 4 + kh];
            a.y = s_pdf[m * NBINS + ch * 4 + kh + 1];
            c = __builtin_amdgcn_wmma_f32_16x16x4_f32(
                    false, a, false, ones, (short)0, c, false, false);
        }
        // D tile: lane 0 holds N=0 rows 0..7, lane 16 holds N=0 rows 8..15
        if (lane == 0)  { for (int r = 0; r < 8; ++r) s_den[r]     = c[r] + EPSV; }
        if (lane == 16) { for (int r = 0; r < 8; ++r) s_den[8 + r] = c[r] + EPSV; }
        // same-wave LDS ops are in-order (DScnt), so the reads below are safe

        // --- entropy: rowsum_b q*log(q), q = pdf/den + eps ---
        float den = s_den[m];
        v8f e = {};
        for (int ch = 0; ch < 16; ++ch) {
            float q0 = s_pdf[m * NBINS + ch * 4 + kh]     / den + EPSV;
            float q1 = s_pdf[m * NBINS + ch * 4 + kh + 1] / den + EPSV;
            v2f a;
            a.x = q0 * __logf(q0);
            a.y = q1 * __logf(q1);
            e = __builtin_amdgcn_wmma_f32_16x16x4_f32(
                    false, a, false, ones, (short)0, e, false, false);
        }
        float* ent = wsf + 1024 + blockIdx.x * 16;
        if (lane == 0)  { for (int r = 0; r < 8; ++r) ent[r]     = -e[r]; }
        if (lane == 16) { for (int r = 0; r < 8; ++r) ent[8 + r] = -e[r]; }
    }
}

// ---------------- per-batch bitonic argsort + mask/ids_restore ----------------
__global__ void __launch_bounds__(512) k_sort_mask(const float* __restrict__ ent,
                                                   float* __restrict__ out,
                                                   int* __restrict__ keep) {
    __shared__ float sv[LSEQ];
    __shared__ int   si[LSEQ];
    int t = threadIdx.x;
    int n = blockIdx.x;
    sv[t] = ent[n * LSEQ + t];
    si[t] = t;                                   // index tiebreak -> stable order
    __syncthreads();

    for (int k = 2; k <= LSEQ; k <<= 1) {
        for (int j = k >> 1; j > 0; j >>= 1) {
            int ixj = t ^ j;
            if (ixj > t) {
                float va = sv[t], vb = sv[ixj];
                int   ia = si[t], ib = si[ixj];
                bool gt = (va > vb) || (va == vb && ia > ib);
                bool up = ((t & k) == 0);
                if (gt == up) { sv[t] = vb; sv[ixj] = va; si[t] = ib; si[ixj] = ia; }
            }
            __syncthreads();
        }
    }

    int idx = si[t];                              // ids_shuffle[t]
    // ids_restore[idx] = t ; mask gathered by ids_restore => mask[idx] = (t>=128)
    out[OUT_MASK_OFF + n * LSEQ + idx] = (t < LEN_KEEP) ? 0.0f : 1.0f;
    out[OUT_IDSR_OFF + n * LSEQ + idx] = (float)t;
    if (t < LEN_KEEP) keep[n * LEN_KEEP + t] = idx;
}

// ---------------- gather kept rows of x (float4 streaming copy) ----------------
__global__ void __launch_bounds__(256) k_gather(const float* __restrict__ x,
                                                const int* __restrict__ keep,
                                                float* __restrict__ out) {
    int b = blockIdx.x;                           // n*128 + j
    int n = b >> 7;
    int l = keep[b];
    const float4* src = (const float4*)(x + ((size_t)(n * LSEQ + l)) * DDIM);
    float4*       dst = (float4*)(out + OUT_XMASK_OFF + (size_t)b * DDIM);
    dst[threadIdx.x] = src[threadIdx.x];
}

extern "C" void kernel_launch(void* const* d_in, const int* in_sizes, int n_in,
                              void* d_out, int out_size, void* d_ws, size_t ws_size,
                              hipStream_t stream) {
    const float* x   = (const float*)d_in[0];     // (4,512,1024)
    const float* img = (const float*)d_in[1];     // (4,512,768)
    float* out = (float*)d_out;
    float* wsf = (float*)d_ws;                    // [0..255] mins, [256..511] maxs,
                                                  // [512..513] global, [1024..3071] entropies
    int*   wsi = (int*)((char*)d_ws + 16384);     // ids_keep[512]

    k_minmax_part <<<128, 256, 0, stream>>>(img, wsf);
    k_minmax_final<<<1,   256, 0, stream>>>(wsf, 128);
    k_entropy     <<<(NBATCH * LSEQ) / 16, 256, 0, stream>>>(img, wsf);
    k_sort_mask   <<<NBATCH, LSEQ, 0, stream>>>(wsf + 1024, out, wsi);
    k_gather      <<<NBATCH * LEN_KEEP, 256, 0, stream>>>(x, wsi, out);
}